// Pose2Mesh_18425409700475
// MI455X (gfx1250) — compile-verified
//
#include <hip/hip_runtime.h>
#include <hip/hip_bf16.h>

// ---------------------------------------------------------------------------
// Pose2Mesh regressor, fully collapsed to out[b] = x[b](57) @ C(57x151) + d.
// C and d are rebuilt on-device each launch (deterministic), then a wave32
// WMMA-f32 streaming GEMM produces all 5 outputs. Bandwidth-bound: ~7 MB of
// traffic total -> ~0.3us at 23.3 TB/s on MI455X.
// ---------------------------------------------------------------------------

typedef __attribute__((ext_vector_type(2))) float v2f;
typedef __attribute__((ext_vector_type(8))) float v8f;

__constant__ int cNEB1[24] = {17,11,12,17,13,14,17,15,16,18,15,16,
                              18,18,18,18, 5, 6, 7, 8, 9,10, 9,10};
__constant__ int cNEB2[24] = {18,17,17,18,11,12,18,13,14,17,13,14,
                               0, 5, 6, 0,18,18, 5, 6, 7, 8, 7, 8};

#define Bsz   8192
#define Jn    19
#define SJ    24
#define Dd    2048
#define Hh    512
#define KDIM  57        // J*3
#define NCOL  151       // 72 pose + 3 cam + 46 phi + 20 leaf + 10 shape
#define KPAD  60
#define NPAD  160

// Output region bases (floats) in d_out, return-order concat:
#define OFF_POSE  0
#define OFF_CAM   (Bsz*72)
#define OFF_PHI   (OFF_CAM + Bsz*3)
#define OFF_LEAF  (OFF_PHI + Bsz*46)
#define OFF_SHAPE (OFF_LEAF + Bsz*20)

// ---------------------------------------------------------------------------
// Kernel 1: per-joint collapse  M1[j] = L1_w[j](3xH) @ R1[j](Hx3), same for 2,
// and pose bias b[j] = L1_b[j]@R1[j] + L2_b[j]@R2[j] + reg_b[j].
// ---------------------------------------------------------------------------
__global__ __launch_bounds__(64) void build_joint_mats(
    const float* __restrict__ L1_w, const float* __restrict__ L1_b,
    const float* __restrict__ L2_w, const float* __restrict__ L2_b,
    const float* __restrict__ R1,   const float* __restrict__ R2,
    const float* __restrict__ reg_b,
    float* __restrict__ M1, float* __restrict__ M2, float* __restrict__ bvec)
{
    const int j = blockIdx.x;
    const int tid = threadIdx.x;
    if (tid < 9) {                                  // M1[j][c][o]
        const int c = tid / 3, o = tid % 3;
        float acc = 0.f;
        for (int h = 0; h < Hh; ++h)
            acc += L1_w[(j*3 + c)*Hh + h] * R1[(j*Hh + h)*3 + o];
        M1[j*9 + tid] = acc;
    } else if (tid < 18) {                          // M2[j][c][o]
        const int t = tid - 9, c = t / 3, o = t % 3;
        float acc = 0.f;
        for (int h = 0; h < Hh; ++h)
            acc += L2_w[(j*3 + c)*Hh + h] * R2[(j*Hh + h)*3 + o];
        M2[j*9 + t] = acc;
    } else if (tid < 21) {                          // bias[j][o]
        const int o = tid - 18;
        float acc = reg_b[j*3 + o];
        for (int h = 0; h < Hh; ++h) {
            acc += L1_b[j*Hh + h] * R1[(j*Hh + h)*3 + o];
            acc += L2_b[j*Hh + h] * R2[(j*Hh + h)*3 + o];
        }
        bvec[j*3 + o] = acc;
    }
}

// ---------------------------------------------------------------------------
// Kernel 2: build fused operator C (KPAD x NPAD, zero padded) and bias d.
// One block per (padded) output column. 35 MFLOP total -> negligible.
// ---------------------------------------------------------------------------
__global__ __launch_bounds__(64) void build_cmat(
    const float* __restrict__ W_enc,  const float* __restrict__ b_enc,
    const float* __restrict__ W_shape,const float* __restrict__ b_shape,
    const float* __restrict__ W_cam,  const float* __restrict__ b_cam,
    const float* __restrict__ W_phi,  const float* __restrict__ b_phi,
    const float* __restrict__ W_leaf, const float* __restrict__ b_leaf,
    const float* __restrict__ init_shape, const float* __restrict__ init_cam,
    const float* __restrict__ Rf,
    const float* __restrict__ M1, const float* __restrict__ M2,
    const float* __restrict__ bvec,
    float* __restrict__ Cmat, float* __restrict__ biasv)
{
    __shared__ float sW[Dd];
    __shared__ float sC[KPAD];
    __shared__ float sBdot;

    const int col = blockIdx.x;
    const int tid = threadIdx.x;

    if (col >= NCOL) {                              // zero padding columns
        if (tid < KPAD) Cmat[tid*NPAD + col] = 0.f;
        if (tid == 0)   biasv[col] = 0.f;
        return;
    }

    // Stage this column of the 2048-wide head operator (root-joint already
    // subtracted for pose columns: Rf[j] - Rf[0]).
    for (int f = tid; f < Dd; f += 64) {
        float w;
        if (col < 72) {
            const int j = col / 3, o = col % 3;
            w = Rf[(j*Dd + f)*3 + o] - Rf[f*3 + o];
        } else if (col < 75)  w = W_cam [f*3  + (col-72)];
        else if  (col < 121)  w = W_phi [f*46 + (col-75)];
        else if  (col < 141)  w = W_leaf[f*20 + (col-121)];
        else                  w = W_shape[f*10 + (col-141)];
        sW[f] = w;
    }
    __syncthreads();

    if (tid < KDIM) {                               // C[k][col] = W_enc[k,:].col
        const float* wr = W_enc + tid*Dd;
        float acc = 0.f;
        for (int f = 0; f < Dd; ++f) acc += wr[f] * sW[f];
        sC[tid] = acc;
    } else if (tid < KPAD) {
        sC[tid] = 0.f;
    } else if (tid == 60) {                         // b_enc feeds every head
        float acc = 0.f;
        for (int f = 0; f < Dd; ++f) acc += b_enc[f] * sW[f];
        sBdot = acc;
    }
    __syncthreads();

    if (tid == 0) {
        float bias = sBdot;
        if (col < 72) {                             // scatter neighbor 3x3s
            const int j = col / 3, o = col % 3;
            for (int c = 0; c < 3; ++c) {
                sC[cNEB1[j]*3 + c] += M1[j*9 + c*3 + o];
                sC[cNEB2[j]*3 + c] += M2[j*9 + c*3 + o];
                sC[cNEB1[0]*3 + c] -= M1[c*3 + o];      // root subtraction
                sC[cNEB2[0]*3 + c] -= M2[c*3 + o];
            }
            bias += bvec[j*3 + o] - bvec[o];
        } else if (col < 75)  bias += b_cam [col-72]  + init_cam[col-72];
        else if  (col < 121)  bias += b_phi [col-75];
        else if  (col < 141)  bias += b_leaf[col-121];
        else                  bias += b_shape[col-141] + init_shape[col-141];
        biasv[col] = bias;
    }
    __syncthreads();

    if (tid < KPAD) Cmat[tid*NPAD + col] = sC[tid];
}

// ---------------------------------------------------------------------------
// Kernel 3: streaming GEMM  out(8192x151) = X(8192x57) @ C + d  via
// V_WMMA_F32_16X16X4_F32 (wave32). 8 waves/WG, 16 batch rows per wave,
// 10 column tiles x 15 K-steps. C + bias live in LDS (39 KB/WG).
// ---------------------------------------------------------------------------
__global__ __launch_bounds__(256) void pose_main(
    const float* __restrict__ joints,
    const float* __restrict__ Cmat,
    const float* __restrict__ biasv,
    float* __restrict__ out)
{
    __shared__ float sC[KPAD * NPAD];
    __shared__ float sBias[NPAD];

    const int tid = threadIdx.x;
    for (int i = tid; i < KPAD * NPAD; i += 256) sC[i] = Cmat[i];
    if (tid < NPAD) sBias[tid] = biasv[tid];
    __syncthreads();

    const int wave = tid >> 5;
    const int lane = tid & 31;
    const int hi   = lane >> 4;        // K half-select (f32 WMMA A/B layout)
    const int ln   = lane & 15;        // M for A, N for B/C/D
    const int m0   = blockIdx.x * 128 + wave * 16;

    // A fragments for all 15 K-steps: lane holds row (m0+ln), elements
    // {4t+2*hi, 4t+1+2*hi}; zero-pad K=57..59.
    float a0[15], a1[15];
    const float* jrow = joints + (size_t)(m0 + ln) * KDIM;
#pragma unroll
    for (int t = 0; t < 15; ++t) {
        const int k0 = 4*t + 2*hi;
        a0[t] = (k0     < KDIM) ? jrow[k0]     : 0.f;
        a1[t] = (k0 + 1 < KDIM) ? jrow[k0 + 1] : 0.f;
    }

    for (int n = 0; n < 10; ++n) {
        const int col = n*16 + ln;
        const float bias = sBias[col];
        v8f acc;
#pragma unroll
        for (int v = 0; v < 8; ++v) acc[v] = bias;   // seed with d

#pragma unroll
        for (int t = 0; t < 15; ++t) {
            const int kb = 4*t + 2*hi;
            v2f av; av[0] = a0[t];               av[1] = a1[t];
            v2f bv; bv[0] = sC[kb*NPAD + col];   bv[1] = sC[(kb+1)*NPAD + col];
            acc = __builtin_amdgcn_wmma_f32_16x16x4_f32(
                    false, av, false, bv, (short)0, acc, false, false);
        }

        if (col < NCOL) {
            // Map column -> (region base, row stride, column-in-region)
            long base; int stride, off;
            if      (col < 72)  { base = OFF_POSE;  stride = 72; off = col;      }
            else if (col < 75)  { base = OFF_CAM;   stride = 3;  off = col-72;   }
            else if (col < 121) { base = OFF_PHI;   stride = 46; off = col-75;   }
            else if (col < 141) { base = OFF_LEAF;  stride = 20; off = col-121;  }
            else                { base = OFF_SHAPE; stride = 10; off = col-141;  }
#pragma unroll
            for (int v = 0; v < 8; ++v) {
                const int m = m0 + v + 8*hi;     // D layout: M = v + 8*(lane/16)
                out[base + (long)m * stride + off] = acc[v];
            }
        }
    }
}

// ---------------------------------------------------------------------------
extern "C" void kernel_launch(void* const* d_in, const int* in_sizes, int n_in,
                              void* d_out, int out_size, void* d_ws, size_t ws_size,
                              hipStream_t stream) {
    (void)in_sizes; (void)n_in; (void)out_size; (void)ws_size;
    const float* joints     = (const float*)d_in[0];
    const float* W_enc      = (const float*)d_in[1];
    const float* b_enc      = (const float*)d_in[2];
    const float* W_shape    = (const float*)d_in[3];
    const float* b_shape    = (const float*)d_in[4];
    const float* W_cam      = (const float*)d_in[5];
    const float* b_cam      = (const float*)d_in[6];
    const float* W_phi      = (const float*)d_in[7];
    const float* b_phi      = (const float*)d_in[8];
    const float* W_leaf     = (const float*)d_in[9];
    const float* b_leaf     = (const float*)d_in[10];
    const float* init_shape = (const float*)d_in[11];
    const float* init_cam   = (const float*)d_in[12];
    const float* L1_w       = (const float*)d_in[13];
    const float* L1_b       = (const float*)d_in[14];
    const float* L2_w       = (const float*)d_in[15];
    const float* L2_b       = (const float*)d_in[16];
    const float* Rf         = (const float*)d_in[17];
    const float* R1         = (const float*)d_in[18];
    const float* R2         = (const float*)d_in[19];
    const float* reg_b      = (const float*)d_in[20];

    float* w     = (float*)d_ws;
    float* M1    = w;                  // 24*9   = 216
    float* M2    = w + 216;            // 216
    float* bvec  = w + 432;            // 72
    float* Cmat  = w + 512;            // 60*160 = 9600
    float* biasv = w + 512 + KPAD*NPAD;// 160     (total ~41 KB)

    build_joint_mats<<<SJ, 64, 0, stream>>>(L1_w, L1_b, L2_w, L2_b,
                                            R1, R2, reg_b, M1, M2, bvec);
    build_cmat<<<NPAD, 64, 0, stream>>>(W_enc, b_enc, W_shape, b_shape,
                                        W_cam, b_cam, W_phi, b_phi,
                                        W_leaf, b_leaf, init_shape, init_cam,
                                        Rf, M1, M2, bvec, Cmat, biasv);
    pose_main<<<Bsz / 128, 256, 0, stream>>>(joints, Cmat, biasv,
                                             (float*)d_out);
}